// Network_28398323761347
// MI455X (gfx1250) — compile-verified
//
#include <hip/hip_runtime.h>
#include <hip/hip_bf16.h>
#include <math.h>

#define NN     50000
#define EE     800000
#define IND    500
#define HIDD   128
#define OUTD   64

typedef __attribute__((ext_vector_type(2))) float v2f;
typedef __attribute__((ext_vector_type(8))) float v8f;
typedef unsigned int u32;
typedef __attribute__((ext_vector_type(4))) unsigned int u32x4;
typedef __attribute__((ext_vector_type(8))) int i32x8;
typedef __attribute__((ext_vector_type(4))) int i32x4;

#if defined(__has_builtin)
#if __has_builtin(__builtin_amdgcn_tensor_load_to_lds) && __has_builtin(__builtin_amdgcn_s_wait_tensorcnt)
#define HAVE_TDM 1
#endif
#endif

// ---- monotonic float<->uint encoding for atomic max on floats ----
__device__ __forceinline__ unsigned encf(float x) {
    unsigned u = __float_as_uint(x);
    return (u & 0x80000000u) ? ~u : (u | 0x80000000u);
}
__device__ __forceinline__ float decf(unsigned u) {
    return (u & 0x80000000u) ? __uint_as_float(u & 0x7fffffffu) : __uint_as_float(~u);
}
#define ENC_NEG_INF 0x007FFFFFu   // encf(-inf)

__device__ __forceinline__ float eluf(float v)   { return v > 0.f ? v : (expf(v) - 1.f); }
__device__ __forceinline__ float lrelu02(float v){ return v > 0.f ? v : 0.2f * v; }

#ifdef HAVE_TDM
// Issue a TDM 2-D tile load (global -> LDS) with per-row LDS padding.
// D# field packing per CDNA5 ISA ch.8 (group0 128b, group1 256b).
// pad_int_code: dwords-before-pad = 2<<code ; pad_amt_code: pad dwords = code+1.
__device__ __forceinline__ void tdm_load_2d(u32 lds_byte_off, const void* gptr,
                                            u32 width, u32 rows, u32 stride,
                                            u32 pad_int_code, u32 pad_amt_code)
{
    const unsigned long long ga = (unsigned long long)(size_t)gptr;
    u32x4 g0;
    g0[0] = 1u;                                            // count=1, user mode, no gather
    g0[1] = lds_byte_off;                                  // lds_addr
    g0[2] = (u32)ga;                                       // global_addr[31:0]
    g0[3] = (u32)((ga >> 32) & 0x01ffffffull) | 0x80000000u; // global_addr[56:32] | type=2
    i32x8 g1;
    g1[0] = (int)((2u << 16) | (1u << 20) |                // data_size=4B, pad_enable
                  (pad_int_code << 22) | (pad_amt_code << 25));
    g1[1] = (int)((width & 0xffffu) << 16);                // tensor_dim0[15:0]
    g1[2] = (int)(((width >> 16) & 0xffffu) |              // tensor_dim0[31:16]
                  ((rows & 0xffffu) << 16));               // tensor_dim1[15:0]
    g1[3] = (int)(((rows >> 16) & 0xffffu) |               // tensor_dim1[31:16]
                  ((width & 0xffffu) << 16));              // tile_dim0 = width
    g1[4] = (int)(rows & 0xffffu);                         // tile_dim1 = rows, tile_dim2 = 0
    g1[5] = (int)stride;                                   // tensor_dim0_stride[31:0]
    g1[6] = 0;                                             // stride[47:32] = 0
    g1[7] = 0;
    i32x4 z4 = {0, 0, 0, 0};
    i32x8 z8 = {0, 0, 0, 0, 0, 0, 0, 0};
    __builtin_amdgcn_tensor_load_to_lds(g0, g1, z4, z4, z8, 0);
}
#endif

// =====================================================================
// WMMA fp32 GEMM: C[M x COLS] = sum_t A_t[M x K] @ W_t[K x COLS] (+bias)(+relu)
// One wave per 16-row strip, full COLS width (COLS/16 accumulators of v8f).
// B is staged into LDS (TDM async-tensor load, padded rows: stride ≡ 8 mod 32
// so the two half-wave row reads land in disjoint bank groups), A streams
// from global. Uses V_WMMA_F32_16X16X4_F32.
// =====================================================================
template<int COLS, int TERMS, bool RELU, bool BIAS>
__global__ __launch_bounds__(128)
void gemm_wmma(const float* __restrict__ A0, const float* __restrict__ W0,
               const float* __restrict__ A1, const float* __restrict__ W1,
               const float* __restrict__ A2, const float* __restrict__ W2,
               const float* __restrict__ bias, float* __restrict__ C,
               int M, int K)
{
    constexpr int NT    = COLS / 16;
    constexpr int PADW  = 8;                 // LDS row padding (dwords)
    constexpr int S     = COLS + PADW;       // LDS row stride; S % 32 == 8
    constexpr int CHUNK = 120;               // rows staged per iteration (mult. of 4)
    constexpr u32 PAD_INT = (COLS == 128) ? 6u : 5u;  // 2<<6=128 / 2<<5=64 dwords
    constexpr u32 PAD_AMT = (u32)(PADW - 1);
    __shared__ float sW[CHUNK * S];

    const int wid  = threadIdx.x >> 5;
    const int lane = threadIdx.x & 31;
    const int row0 = (blockIdx.x * 4 + wid) * 16;
    const bool active = row0 < M;            // tail waves still run barriers
    const int half = lane >> 4;              // 0: K+0..1 lanes, 1: K+2..3 lanes
    const int r    = lane & 15;

    v8f acc[NT] = {};
    const float* As[3] = {A0, A1, A2};
    const float* Ws[3] = {W0, W1, W2};

#pragma unroll
    for (int term = 0; term < TERMS; ++term) {
        const float* __restrict__ Wg = Ws[term];
        const float* __restrict__ arow =
            As[term] + (size_t)((active ? row0 : 0) + r) * K + 2 * half;

        for (int k0 = 0; k0 < K; k0 += CHUNK) {
            const int rows = (K - k0 < CHUNK) ? (K - k0) : CHUNK;
#ifdef HAVE_TDM
            if (wid == 0) {
                tdm_load_2d(0u, Wg + (size_t)k0 * COLS, (u32)COLS, (u32)rows,
                            (u32)COLS, PAD_INT, PAD_AMT);
                __builtin_amdgcn_s_wait_tensorcnt(0);
            }
#else
            for (int idx = threadIdx.x; idx < rows * COLS; idx += 128) {
                const int rr = idx / COLS, cc = idx % COLS;
                sW[rr * S + cc] = Wg[(size_t)(k0 + rr) * COLS + cc];
            }
#endif
            __syncthreads();

            if (active) {
                for (int kk = 0; kk < rows; kk += 4) {
                    v2f av;
                    av.x = arow[k0 + kk];
                    av.y = arow[k0 + kk + 1];
                    const float* ws = sW + (kk + 2 * half) * S + r;
#pragma unroll
                    for (int t = 0; t < NT; ++t) {
                        v2f bv;
                        bv.x = ws[t * 16];
                        bv.y = ws[t * 16 + S];
                        acc[t] = __builtin_amdgcn_wmma_f32_16x16x4_f32(
                            false, av, false, bv, (short)0, acc[t], false, false);
                    }
                }
            }
            __syncthreads();
        }
    }

    if (!active) return;
#pragma unroll
    for (int t = 0; t < NT; ++t) {
#pragma unroll
        for (int v = 0; v < 8; ++v) {
            const int row = row0 + half * 8 + v;   // C/D layout: lanes16-31 hold M=8..15
            const int col = t * 16 + r;
            float val = acc[t][v];
            if (BIAS) val += bias[col];
            if (RELU) val = fmaxf(val, 0.f);
            C[(size_t)row * COLS + col] = val;
        }
    }
}

// =====================================================================
// helpers / elementwise kernels
// =====================================================================
__global__ void fill_f32(float* p, float v, size_t n) {
    size_t i = (size_t)blockIdx.x * blockDim.x + threadIdx.x;
    size_t st = (size_t)gridDim.x * blockDim.x;
    for (; i < n; i += st) p[i] = v;
}

__global__ void degree_k(const int* __restrict__ dst, float* cnt, int E) {
    int e = blockIdx.x * blockDim.x + threadIdx.x;
    if (e < E) atomicAdd(cnt + dst[e], 1.0f);
}

__global__ void init_layer_vec(float* agg_sum, unsigned* agg_max,
                               float* acc_gcn, float* acc_gat, size_t n) {
    size_t i = (size_t)blockIdx.x * blockDim.x + threadIdx.x;
    if (i >= n) return;
    agg_sum[i] = 0.f; acc_gcn[i] = 0.f; acc_gat[i] = 0.f;
    agg_max[i] = ENC_NEG_INF;
}

__global__ void init_layer_node(unsigned* emax, float* denom, int n) {
    int i = blockIdx.x * blockDim.x + threadIdx.x;
    if (i >= n) return;
    emax[i] = ENC_NEG_INF; denom[i] = 0.f;
}

// wave-per-edge: sum/max aggregation of h[src] + GCN-normalized scatter of hg[src]
__global__ __launch_bounds__(256)
void edge_pre(const int* __restrict__ src, const int* __restrict__ dst,
              const float* __restrict__ h, const float* __restrict__ hg,
              const float* __restrict__ cnt,
              float* agg_sum, unsigned* agg_max, float* acc_gcn, int E)
{
    const int wid = threadIdx.x >> 5, lane = threadIdx.x & 31;
    const long e = (long)blockIdx.x * 8 + wid;
    if (e >= E) return;
    const int s = src[e], d = dst[e];
    const float nrm = rsqrtf((cnt[s] + 1.f) * (cnt[d] + 1.f));
    const int c0 = lane * 4;
    const float4 hv = *(const float4*)(h  + (size_t)s * HIDD + c0);
    const float4 gv = *(const float4*)(hg + (size_t)s * HIDD + c0);
    float*    as = agg_sum + (size_t)d * HIDD + c0;
    unsigned* am = agg_max + (size_t)d * HIDD + c0;
    float*    ag = acc_gcn + (size_t)d * HIDD + c0;
    atomicAdd(as + 0, hv.x); atomicAdd(as + 1, hv.y);
    atomicAdd(as + 2, hv.z); atomicAdd(as + 3, hv.w);
    atomicMax(am + 0, encf(hv.x)); atomicMax(am + 1, encf(hv.y));
    atomicMax(am + 2, encf(hv.z)); atomicMax(am + 3, encf(hv.w));
    atomicAdd(ag + 0, nrm * gv.x); atomicAdd(ag + 1, nrm * gv.y);
    atomicAdd(ag + 2, nrm * gv.z); atomicAdd(ag + 3, nrm * gv.w);
}

// wave-per-node: e_src = hw[n].a0 , e_dst = hw[n].a1  (128-wide dots)
__global__ __launch_bounds__(256)
void gat_dots(const float* __restrict__ hw, const float* __restrict__ a0,
              const float* __restrict__ a1, float* e_src, float* e_dst, int N)
{
    const int wid = threadIdx.x >> 5, lane = threadIdx.x & 31;
    const long n = (long)blockIdx.x * 8 + wid;
    if (n >= N) return;
    const int c0 = lane * 4;
    const float4 v  = *(const float4*)(hw + (size_t)n * HIDD + c0);
    const float4 w0 = *(const float4*)(a0 + c0);
    const float4 w1 = *(const float4*)(a1 + c0);
    float p0 = v.x * w0.x + v.y * w0.y + v.z * w0.z + v.w * w0.w;
    float p1 = v.x * w1.x + v.y * w1.y + v.z * w1.z + v.w * w1.w;
#pragma unroll
    for (int o = 16; o >= 1; o >>= 1) {
        p0 += __shfl_xor(p0, o, 32);
        p1 += __shfl_xor(p1, o, 32);
    }
    if (lane == 0) { e_src[n] = p0; e_dst[n] = p1; }
}

__global__ void gat_e1(const int* __restrict__ src, const int* __restrict__ dst,
                       const float* __restrict__ es, const float* __restrict__ ed,
                       float* ebuf, unsigned* emax, int E)
{
    int e = blockIdx.x * blockDim.x + threadIdx.x;
    if (e >= E) return;
    const float v = lrelu02(es[src[e]] + ed[dst[e]]);
    ebuf[e] = v;
    atomicMax(emax + dst[e], encf(v));
}

__global__ void gat_e2(const int* __restrict__ dst, float* ebuf,
                       const unsigned* __restrict__ emax, float* denom, int E)
{
    int e = blockIdx.x * blockDim.x + threadIdx.x;
    if (e >= E) return;
    float m = decf(emax[dst[e]]);
    if (!isfinite(m)) m = 0.f;
    const float ee = expf(ebuf[e] - m);
    ebuf[e] = ee;
    atomicAdd(denom + dst[e], ee);
}

// wave-per-edge: acc_gat[dst] += coef * hw[src]
__global__ __launch_bounds__(256)
void gat_scatter(const int* __restrict__ src, const int* __restrict__ dst,
                 const float* __restrict__ hw, const float* __restrict__ ebuf,
                 const float* __restrict__ denom, float* acc_gat, int E)
{
    const int wid = threadIdx.x >> 5, lane = threadIdx.x & 31;
    const long e = (long)blockIdx.x * 8 + wid;
    if (e >= E) return;
    const int s = src[e], d = dst[e];
    const float coef = ebuf[e] / (denom[d] + 1e-16f);
    const int c0 = lane * 4;
    const float4 wv = *(const float4*)(hw + (size_t)s * HIDD + c0);
    float* ag = acc_gat + (size_t)d * HIDD + c0;
    atomicAdd(ag + 0, coef * wv.x); atomicAdd(ag + 1, coef * wv.y);
    atomicAdd(ag + 2, coef * wv.z); atomicAdd(ag + 3, coef * wv.w);
}

// mean_agg = agg_sum / max(cnt,1) ; decode agg_max in place (non-finite -> 0)
__global__ void prep_k(const float* __restrict__ agg_sum, unsigned* agg_max,
                       const float* __restrict__ cnt, float* mean_agg, size_t total)
{
    size_t idx = (size_t)blockIdx.x * blockDim.x + threadIdx.x;
    if (idx >= total) return;
    const int n = (int)(idx >> 7);
    mean_agg[idx] = agg_sum[idx] * (1.f / fmaxf(cnt[n], 1.f));
    float m = decf(agg_max[idx]);
    if (!isfinite(m)) m = 0.f;
    ((float*)agg_max)[idx] = m;
}

// out = sum_k softmax(alphas)[k] * elu(o_k)  ; o_gcn assembled inline
__global__ void na_combine(const float* __restrict__ hg, const float* __restrict__ acc_gcn,
                           const float* __restrict__ o_sm, const float* __restrict__ o_sx,
                           const float* __restrict__ o_gin, const float* __restrict__ acc_gat,
                           const float* __restrict__ cnt, const float* __restrict__ alphas,
                           float* out, size_t total)
{
    size_t idx = (size_t)blockIdx.x * blockDim.x + threadIdx.x;
    if (idx >= total) return;
    float a[5], w[5];
    float mx = -3.4e38f;
#pragma unroll
    for (int k = 0; k < 5; ++k) { a[k] = alphas[k]; mx = fmaxf(mx, a[k]); }
    float s = 0.f;
#pragma unroll
    for (int k = 0; k < 5; ++k) { w[k] = expf(a[k] - mx); s += w[k]; }
    const float inv = 1.f / s;
    const int n = (int)(idx >> 7);
    const float invdeg = 1.f / (cnt[n] + 1.f);
    const float o0 = acc_gcn[idx] + hg[idx] * invdeg;
    out[idx] = inv * (w[0] * eluf(o0)        + w[1] * eluf(o_sm[idx]) +
                      w[2] * eluf(o_sx[idx]) + w[3] * eluf(o_gin[idx]) +
                      w[4] * eluf(acc_gat[idx]));
}

// s1, s2, element max and mean of {x3, s1, s2}
__global__ void final_prep(const float* __restrict__ x1, const float* __restrict__ x2,
                           const float* __restrict__ x3, const float* __restrict__ sc,
                           float* s1, float* s2, float* mmax, float* mmean, size_t total)
{
    size_t idx = (size_t)blockIdx.x * blockDim.x + threadIdx.x;
    if (idx >= total) return;
    const float w0 = 1.f / (1.f + expf(sc[0] - sc[1]));  // softmax(sc_alphas[0])[1]
    const float w1 = 1.f / (1.f + expf(sc[2] - sc[3]));  // softmax(sc_alphas[1])[1]
    const float a = w0 * x1[idx], b = w1 * x2[idx], c = x3[idx];
    s1[idx] = a; s2[idx] = b;
    mmax[idx]  = fmaxf(c, fmaxf(a, b));
    mmean[idx] = (a + b + c) * (1.f / 3.f);
}

__global__ void la_combine(const float* __restrict__ o_max, const float* __restrict__ o_cat,
                           const float* __restrict__ o_mean, const float* __restrict__ la,
                           float* out, size_t total)
{
    size_t idx = (size_t)blockIdx.x * blockDim.x + threadIdx.x;
    if (idx >= total) return;
    float a0 = la[0], a1 = la[1], a2 = la[2];
    float mx = fmaxf(a0, fmaxf(a1, a2));
    float e0 = expf(a0 - mx), e1 = expf(a1 - mx), e2 = expf(a2 - mx);
    float inv = 1.f / (e0 + e1 + e2);
    out[idx] = inv * (e0 * o_max[idx] + e1 * o_cat[idx] + e2 * o_mean[idx]);
}

// =====================================================================
extern "C" void kernel_launch(void* const* d_in, const int* in_sizes, int n_in,
                              void* d_out, int out_size, void* d_ws, size_t ws_size,
                              hipStream_t stream)
{
    (void)in_sizes; (void)n_in; (void)out_size; (void)ws_size;
    const float* x        = (const float*)d_in[0];
    const int*   eidx     = (const int*)  d_in[1];
    const float* lin1_w   = (const float*)d_in[2];
    const float* lin1_b   = (const float*)d_in[3];
    const float* gcn_w    = (const float*)d_in[4];
    const float* sm_w1    = (const float*)d_in[5];
    const float* sm_w2    = (const float*)d_in[6];
    const float* sx_w1    = (const float*)d_in[7];
    const float* sx_w2    = (const float*)d_in[8];
    const float* gin_w    = (const float*)d_in[9];
    const float* gat_w    = (const float*)d_in[10];
    const float* gat_a    = (const float*)d_in[11];
    const float* la_max_w = (const float*)d_in[12];
    const float* la_cat_w = (const float*)d_in[13];
    const float* la_mean_w= (const float*)d_in[14];
    const float* clf_w    = (const float*)d_in[15];
    const float* clf_b    = (const float*)d_in[16];
    const float* na_alphas= (const float*)d_in[17];
    const float* sc_alphas= (const float*)d_in[18];
    const float* la_alphas= (const float*)d_in[19];
    float* out = (float*)d_out;

    const int* src = eidx;
    const int* dst = eidx + EE;

    float* ws = (float*)d_ws;
    const size_t NF = (size_t)NN * HIDD;
    float*    h0       = ws;
    float*    x1       = ws + 1 * NF;
    float*    x2       = ws + 2 * NF;
    float*    x3       = ws + 3 * NF;
    float*    hg       = ws + 4 * NF;
    float*    hw       = ws + 5 * NF;
    float*    agg_sum  = ws + 6 * NF;
    unsigned* agg_max  = (unsigned*)(ws + 7 * NF);
    float*    mean_agg = ws + 8 * NF;
    float*    acc_gcn  = ws + 9 * NF;
    float*    acc_gat  = ws + 10 * NF;
    float*    o_sm     = ws + 11 * NF;
    float*    o_sx     = ws + 12 * NF;
    float*    o_gin    = ws + 13 * NF;
    float*    cnt      = ws + 14 * NF;
    float*    e_srcv   = cnt + NN;
    float*    e_dstv   = e_srcv + NN;
    unsigned* emax     = (unsigned*)(e_dstv + NN);
    float*    denom    = e_dstv + 2 * NN;
    float*    ebuf     = denom + NN;

    const int GEMM_GRID = (NN + 63) / 64;         // 4 waves/block, 16 rows/wave
    const int VEC_GRID  = (int)((NF + 255) / 256);
    const int EW_GRID   = (EE + 7) / 8;           // wave-per-edge
    const int NW_GRID   = (NN + 7) / 8;           // wave-per-node
    const int ES_GRID   = (EE + 255) / 256;

    // degree (in-degree over dst), computed once
    fill_f32<<<(NN + 255) / 256, 256, 0, stream>>>(cnt, 0.f, NN);
    degree_k<<<ES_GRID, 256, 0, stream>>>(dst, cnt, EE);

    // h0 = x @ lin1_w + lin1_b
    gemm_wmma<HIDD, 1, false, true><<<GEMM_GRID, 128, 0, stream>>>(
        x, lin1_w, nullptr, nullptr, nullptr, nullptr, lin1_b, h0, NN, IND);

    const float* hin = h0;
    float* houts[3] = {x1, x2, x3};
    for (int i = 0; i < 3; ++i) {
        const size_t wo = (size_t)i * HIDD * HIDD;
        const float* a0 = gat_a + (size_t)i * 2 * HIDD;
        const float* a1 = a0 + HIDD;

        gemm_wmma<HIDD, 1, false, false><<<GEMM_GRID, 128, 0, stream>>>(
            hin, gcn_w + wo, nullptr, nullptr, nullptr, nullptr, nullptr, hg, NN, HIDD);
        gemm_wmma<HIDD, 1, false, false><<<GEMM_GRID, 128, 0, stream>>>(
            hin, gat_w + wo, nullptr, nullptr, nullptr, nullptr, nullptr, hw, NN, HIDD);

        init_layer_vec<<<VEC_GRID, 256, 0, stream>>>(agg_sum, agg_max, acc_gcn, acc_gat, NF);
        init_layer_node<<<(NN + 255) / 256, 256, 0, stream>>>(emax, denom, NN);

        edge_pre<<<EW_GRID, 256, 0, stream>>>(src, dst, hin, hg, cnt,
                                              agg_sum, agg_max, acc_gcn, EE);
        gat_dots<<<NW_GRID, 256, 0, stream>>>(hw, a0, a1, e_srcv, e_dstv, NN);
        gat_e1<<<ES_GRID, 256, 0, stream>>>(src, dst, e_srcv, e_dstv, ebuf, emax, EE);
        gat_e2<<<ES_GRID, 256, 0, stream>>>(dst, ebuf, emax, denom, EE);
        gat_scatter<<<EW_GRID, 256, 0, stream>>>(src, dst, hw, ebuf, denom, acc_gat, EE);

        prep_k<<<VEC_GRID, 256, 0, stream>>>(agg_sum, agg_max, cnt, mean_agg, NF);

        gemm_wmma<HIDD, 2, false, false><<<GEMM_GRID, 128, 0, stream>>>(
            hin, sm_w1 + wo, mean_agg, sm_w2 + wo, nullptr, nullptr, nullptr, o_sm, NN, HIDD);
        gemm_wmma<HIDD, 2, false, false><<<GEMM_GRID, 128, 0, stream>>>(
            hin, sx_w1 + wo, (const float*)agg_max, sx_w2 + wo, nullptr, nullptr, nullptr, o_sx, NN, HIDD);
        gemm_wmma<HIDD, 2, false, false><<<GEMM_GRID, 128, 0, stream>>>(
            hin, gin_w + wo, agg_sum, gin_w + wo, nullptr, nullptr, nullptr, o_gin, NN, HIDD);

        na_combine<<<VEC_GRID, 256, 0, stream>>>(hg, acc_gcn, o_sm, o_sx, o_gin, acc_gat,
                                                 cnt, na_alphas + (size_t)i * 5, houts[i], NF);
        hin = houts[i];
    }

    // final stage: skip-scaled s1/s2 (reuse hg/hw), max -> agg_sum, mean -> mean_agg
    final_prep<<<VEC_GRID, 256, 0, stream>>>(x1, x2, x3, sc_alphas,
                                             hg, hw, agg_sum, mean_agg, NF);

    gemm_wmma<HIDD, 1, true, false><<<GEMM_GRID, 128, 0, stream>>>(
        agg_sum, la_max_w, nullptr, nullptr, nullptr, nullptr, nullptr, o_sm, NN, HIDD);
    gemm_wmma<HIDD, 3, true, false><<<GEMM_GRID, 128, 0, stream>>>(
        x3, la_cat_w, hg, la_cat_w + (size_t)HIDD * HIDD,
        hw, la_cat_w + (size_t)2 * HIDD * HIDD, nullptr, o_sx, NN, HIDD);
    gemm_wmma<HIDD, 1, true, false><<<GEMM_GRID, 128, 0, stream>>>(
        mean_agg, la_mean_w, nullptr, nullptr, nullptr, nullptr, nullptr, o_gin, NN, HIDD);

    la_combine<<<VEC_GRID, 256, 0, stream>>>(o_sm, o_sx, o_gin, la_alphas, acc_gcn, NF);

    // classifier: out = x5 @ clf_w + clf_b   (COLS = 64)
    gemm_wmma<OUTD, 1, false, true><<<GEMM_GRID, 128, 0, stream>>>(
        acc_gcn, clf_w, nullptr, nullptr, nullptr, nullptr, clf_b, out, NN, HIDD);
}